// DifferentiableSuperpixelTokenizer_4063039062509
// MI455X (gfx1250) — compile-verified
//
#include <hip/hip_runtime.h>

// ---------------------------------------------------------------------------
// DifferentiableSuperpixelTokenizer for MI455X (gfx1250, wave32)
//
// out[b,s,:] = ((mean_s(seg_mean[b,s,:]) @ W_proj + b_proj) @ W_gcn + b_gcn)
// (identical for all s: the reference mean-pools over nodes and broadcasts).
// HBM-bound: 268 MB pixel scatter + 50 MB broadcast store (~14 us floor at
// 23.3 TB/s). GEMM chain uses native fp32 WMMA (V_WMMA_F32_16X16X4_F32),
// K-split across 8 waves/block with LDS reduction for latency hiding.
// ---------------------------------------------------------------------------

typedef __attribute__((ext_vector_type(2))) float v2f;
typedef __attribute__((ext_vector_type(8))) float v8f;

#define BATCH 64
#define HWPIX (512 * 512)
#define SEGS  256
#define EDIM  768
#define NREP  4       // LDS bin replicas to spread atomic contention
#define CHUNK 8192    // pixels per block in the scatter phase
#define KWAVES 8      // waves per block in gemm2 (K-split)

// ---------------------------------------------------------------------------
// 0) zero the global segment accumulator [B, SEGS, 4] (sumR,sumG,sumB,count)
// ---------------------------------------------------------------------------
__global__ void __launch_bounds__(256) zero_ws_kernel(float4* __restrict__ p, int n4) {
  int i = blockIdx.x * blockDim.x + threadIdx.x;
  if (i < n4) p[i] = make_float4(0.f, 0.f, 0.f, 0.f);
}

// ---------------------------------------------------------------------------
// 1) segment-sum scatter: img [B,3,H,W] + segments [B,H,W] -> accum [B,S,4]
//    b128 loads, replicated LDS f32 atomics, global f32 atomic flush.
// ---------------------------------------------------------------------------
__global__ void __launch_bounds__(256) segsum_kernel(const float* __restrict__ img,
                                                     const int*   __restrict__ seg,
                                                     float*       __restrict__ accum) {
  __shared__ float bins[NREP * SEGS * 4];   // 16 KB
  const int tid   = threadIdx.x;
  const int chunk = blockIdx.x;
  const int b     = blockIdx.y;

  for (int i = tid; i < NREP * SEGS * 4; i += 256) bins[i] = 0.f;
  __syncthreads();

  const float4* pr = (const float4*)(img + (size_t)b * 3 * HWPIX);
  const float4* pg = (const float4*)(img + (size_t)b * 3 * HWPIX + HWPIX);
  const float4* pb = (const float4*)(img + (size_t)b * 3 * HWPIX + 2 * HWPIX);
  const int4*   ps = (const int4*)(seg + (size_t)b * HWPIX);

  const int rep = (tid & (NREP - 1)) * (SEGS * 4);
  int q = (chunk * CHUNK) / 4 + tid;          // float4 index into this image
#pragma unroll 2
  for (int it = 0; it < (CHUNK / 4) / 256; ++it, q += 256) {
    float4 r = pr[q], g = pg[q], bl = pb[q];
    int4 s = ps[q];
    atomicAdd(&bins[rep + s.x * 4 + 0], r.x);
    atomicAdd(&bins[rep + s.x * 4 + 1], g.x);
    atomicAdd(&bins[rep + s.x * 4 + 2], bl.x);
    atomicAdd(&bins[rep + s.x * 4 + 3], 1.f);
    atomicAdd(&bins[rep + s.y * 4 + 0], r.y);
    atomicAdd(&bins[rep + s.y * 4 + 1], g.y);
    atomicAdd(&bins[rep + s.y * 4 + 2], bl.y);
    atomicAdd(&bins[rep + s.y * 4 + 3], 1.f);
    atomicAdd(&bins[rep + s.z * 4 + 0], r.z);
    atomicAdd(&bins[rep + s.z * 4 + 1], g.z);
    atomicAdd(&bins[rep + s.z * 4 + 2], bl.z);
    atomicAdd(&bins[rep + s.z * 4 + 3], 1.f);
    atomicAdd(&bins[rep + s.w * 4 + 0], r.w);
    atomicAdd(&bins[rep + s.w * 4 + 1], g.w);
    atomicAdd(&bins[rep + s.w * 4 + 2], bl.w);
    atomicAdd(&bins[rep + s.w * 4 + 3], 1.f);
  }
  __syncthreads();

  // each thread owns one segment; fold replicas, flush with global atomics
  float a0 = 0.f, a1 = 0.f, a2 = 0.f, a3 = 0.f;
#pragma unroll
  for (int rp = 0; rp < NREP; ++rp) {
    int o = rp * SEGS * 4 + tid * 4;
    a0 += bins[o + 0]; a1 += bins[o + 1]; a2 += bins[o + 2]; a3 += bins[o + 3];
  }
  float* dst = accum + ((size_t)b * SEGS + tid) * 4;
  atomicAdd(dst + 0, a0);
  atomicAdd(dst + 1, a1);
  atomicAdd(dst + 2, a2);
  atomicAdd(dst + 3, a3);
}

// ---------------------------------------------------------------------------
// 2) per-batch mean over segment means:  Mmat[b, 0..2] (col 3 zero-padded)
// ---------------------------------------------------------------------------
__global__ void __launch_bounds__(256) segmean_kernel(const float* __restrict__ accum,
                                                      float*       __restrict__ Mmat) {
  __shared__ float sr[SEGS], sg[SEGS], sb[SEGS];
  const int b = blockIdx.x, s = threadIdx.x;
  const float* a = accum + ((size_t)b * SEGS + s) * 4;
  const float cnt = fmaxf(a[3], 1.f);     // clip(counts, 1, None)
  sr[s] = a[0] / cnt;
  sg[s] = a[1] / cnt;
  sb[s] = a[2] / cnt;
  __syncthreads();
  for (int off = SEGS / 2; off > 0; off >>= 1) {
    if (s < off) { sr[s] += sr[s + off]; sg[s] += sg[s + off]; sb[s] += sb[s + off]; }
    __syncthreads();
  }
  if (s == 0) {
    Mmat[b * 4 + 0] = sr[0] * (1.f / SEGS);
    Mmat[b * 4 + 1] = sg[0] * (1.f / SEGS);
    Mmat[b * 4 + 2] = sb[0] * (1.f / SEGS);
    Mmat[b * 4 + 3] = 0.f;                // K padding for the 16x16x4 WMMA
  }
}

// ---------------------------------------------------------------------------
// 3) P = Mmat(64x4) @ W_proj(3x768, K padded) + b_proj  -- one WMMA per tile
//    A 16x4 f32 layout (ISA 7.12.2): lanes 0-15 hold K0/K1 in the VGPR pair,
//    lanes 16-31 hold K2/K3. B row-striped across lanes, mirrored split.
// ---------------------------------------------------------------------------
__global__ void __launch_bounds__(32) gemm1_wmma_kernel(const float* __restrict__ Mmat,
                                                        const float* __restrict__ Wp,
                                                        const float* __restrict__ bp,
                                                        float*       __restrict__ P) {
  const int lane = threadIdx.x;
  const int n0 = blockIdx.x * 16;
  const int m0 = blockIdx.y * 16;
  const int rc = lane & 15;     // row for A, col for B/D
  const int kh = lane >> 4;     // K half: 0 -> K0,K1 ; 1 -> K2,K3

  v2f a, bm;
  a.x = Mmat[(m0 + rc) * 4 + kh * 2 + 0];
  a.y = Mmat[(m0 + rc) * 4 + kh * 2 + 1];          // K=3 column is zero
  const int k0 = kh * 2 + 0, k1 = kh * 2 + 1;
  bm.x = (k0 < 3) ? Wp[k0 * EDIM + n0 + rc] : 0.f; // W_proj has only 3 rows
  bm.y = (k1 < 3) ? Wp[k1 * EDIM + n0 + rc] : 0.f;

  v8f c = {};
  c = __builtin_amdgcn_wmma_f32_16x16x4_f32(false, a, false, bm, (short)0, c,
                                            false, false);
  const float bias = bp[n0 + rc];
#pragma unroll
  for (int r = 0; r < 8; ++r) {
    const int m = r + (kh << 3);                   // D: lanes16-31 hold M=r+8
    P[(size_t)(m0 + m) * EDIM + n0 + rc] = c[r] + bias;
  }
}

// ---------------------------------------------------------------------------
// 4) Y = P(64x768) @ W_gcn(768x768) + b_gcn
//    K split across 8 waves per block (96-wide slices, 24 WMMAs each) for
//    occupancy/latency hiding; dual accumulators break the C-chain; the 8
//    partial 16x16 tiles are reduced through LDS.
// ---------------------------------------------------------------------------
__global__ void __launch_bounds__(KWAVES * 32) gemm2_wmma_kernel(
    const float* __restrict__ P,
    const float* __restrict__ Wg,
    const float* __restrict__ bg,
    float*       __restrict__ Y) {
  __shared__ float red[KWAVES * 256];     // 8 KB: one 16x16 partial per wave

  const int tid  = threadIdx.x;
  const int wave = tid >> 5;
  const int lane = tid & 31;
  const int n0 = blockIdx.x * 16;
  const int m0 = blockIdx.y * 16;
  const int rc = lane & 15;
  const int kh = lane >> 4;

  const int kbase = wave * (EDIM / KWAVES);        // 96-wide K slice
  v8f c0 = {}, c1 = {};
#pragma unroll 2
  for (int i = 0; i < EDIM / KWAVES; i += 8) {
    // even K-step -> c0
    {
      const int ka = kbase + i + kh * 2;
      v2f a, bm;
      a.x  = P[(size_t)(m0 + rc) * EDIM + ka + 0];
      a.y  = P[(size_t)(m0 + rc) * EDIM + ka + 1];
      bm.x = Wg[(size_t)(ka + 0) * EDIM + n0 + rc];
      bm.y = Wg[(size_t)(ka + 1) * EDIM + n0 + rc];
      c0 = __builtin_amdgcn_wmma_f32_16x16x4_f32(false, a, false, bm, (short)0,
                                                 c0, false, false);
    }
    // odd K-step -> c1 (independent accumulator, pipelines with c0)
    {
      const int ka = kbase + i + 4 + kh * 2;
      v2f a, bm;
      a.x  = P[(size_t)(m0 + rc) * EDIM + ka + 0];
      a.y  = P[(size_t)(m0 + rc) * EDIM + ka + 1];
      bm.x = Wg[(size_t)(ka + 0) * EDIM + n0 + rc];
      bm.y = Wg[(size_t)(ka + 1) * EDIM + n0 + rc];
      c1 = __builtin_amdgcn_wmma_f32_16x16x4_f32(false, a, false, bm, (short)0,
                                                 c1, false, false);
    }
  }

  // stash this wave's partial tile: slot t = r*32 + lane, per-wave bank
#pragma unroll
  for (int r = 0; r < 8; ++r)
    red[wave * 256 + r * 32 + lane] = c0[r] + c1[r];
  __syncthreads();

  // 256 threads: each owns one of the 256 tile positions; fold 8 waves
  float s = 0.f;
#pragma unroll
  for (int w = 0; w < KWAVES; ++w) s += red[w * 256 + tid];
  const int r = tid >> 5;
  const int l = tid & 31;
  const int m = r + ((l >> 4) << 3);               // D-layout position -> (m,n)
  const int n = l & 15;
  Y[(size_t)(m0 + m) * EDIM + n0 + n] = s + bg[n0 + n];
}

// ---------------------------------------------------------------------------
// 5) broadcast Y[b,:] to out[b, s, :] for all 256 s  (coalesced b128 stores;
//    Y is 196 KB -> L2 resident, write-bound at ~50 MB)
// ---------------------------------------------------------------------------
__global__ void __launch_bounds__(256) broadcast_kernel(const float4* __restrict__ Y4,
                                                        float4*       __restrict__ out4,
                                                        int n4) {
  const int idx = blockIdx.x * blockDim.x + threadIdx.x;
  if (idx >= n4) return;
  const int e4 = idx % (EDIM / 4);
  const int b  = idx / (SEGS * (EDIM / 4));
  out4[idx] = Y4[b * (EDIM / 4) + e4];
}

// ---------------------------------------------------------------------------
extern "C" void kernel_launch(void* const* d_in, const int* in_sizes, int n_in,
                              void* d_out, int out_size, void* d_ws, size_t ws_size,
                              hipStream_t stream) {
  (void)in_sizes; (void)n_in; (void)out_size; (void)ws_size;

  const float* img = (const float*)d_in[0];   // [64,3,512,512] f32
  const int*   seg = (const int*)  d_in[1];   // [64,512,512]   i32
  const float* Wp  = (const float*)d_in[2];   // [3,768]
  const float* bp  = (const float*)d_in[3];   // [768]
  const float* Wg  = (const float*)d_in[4];   // [768,768]
  const float* bg  = (const float*)d_in[5];   // [768]
  float*       out = (float*)d_out;           // [64,256,768]

  // workspace layout (floats): accum | Mmat | P | Y  (~0.66 MB total)
  float* accum = (float*)d_ws;                       // 64*256*4
  float* Mmat  = accum + BATCH * SEGS * 4;           // 64*4
  float* P     = Mmat + BATCH * 4;                   // 64*768
  float* Y     = P + BATCH * EDIM;                   // 64*768

  const int n4a = (BATCH * SEGS * 4) / 4;
  zero_ws_kernel<<<(n4a + 255) / 256, 256, 0, stream>>>((float4*)accum, n4a);

  segsum_kernel<<<dim3(HWPIX / CHUNK, BATCH), 256, 0, stream>>>(img, seg, accum);

  segmean_kernel<<<BATCH, 256, 0, stream>>>(accum, Mmat);

  gemm1_wmma_kernel<<<dim3(EDIM / 16, BATCH / 16), 32, 0, stream>>>(Mmat, Wp, bp, P);

  gemm2_wmma_kernel<<<dim3(EDIM / 16, BATCH / 16), KWAVES * 32, 0, stream>>>(P, Wg, bg, Y);

  const int n4o = BATCH * SEGS * (EDIM / 4);
  broadcast_kernel<<<(n4o + 255) / 256, 256, 0, stream>>>((const float4*)Y,
                                                          (float4*)out, n4o);
}